// MainNetwork_88510686036247
// MI455X (gfx1250) — compile-verified
//
#include <hip/hip_runtime.h>
#include <math.h>

typedef __attribute__((ext_vector_type(2))) float v2f;
typedef __attribute__((ext_vector_type(8))) float v8f;

#define IN_F 2
#define HID 36
#define PAD 48            // 36 padded to 3x16
#define KSTEPS 12         // 48 / 4 (K=4 per fp32 WMMA)
#define WAVES 8
#define ROWS_PER_BLOCK (WAVES * 16)
#define NPARAMS 2809
#define STG_STRIDE 50     // padded row stride for the transpose staging buffer

// parameter layout offsets (row-major W[out][in], then bias)
#define OFF_W1 0
#define OFF_B1 72
#define OFF_W2 108
#define OFF_B2 1404
#define OFF_W3 1440
#define OFF_B3 2736
#define OFF_W4 2772
#define OFF_B4 2808

// Hardware sin: one v_mul (1/2pi) + v_sin_f32 (TRANS unit, co-executes with
// VALU/WMMA). Pre-activations here are O(1), well inside the accurate domain.
__device__ __forceinline__ float hw_sin(float x) { return __sinf(x); }

// One hidden layer: D(16x48) = A(16x48) x B(48x48) via fp32 WMMA, bias + sin,
// then re-layout C/D fragments -> A fragments through per-wave LDS staging.
__device__ __forceinline__ void hidden_layer_wmma(float (&h)[24],
                                                  const v2f* __restrict__ bp,
                                                  const float* __restrict__ bias,
                                                  float* __restrict__ stg,
                                                  int half, int l16)
{
    v8f c0 = {}; v8f c1 = {}; v8f c2 = {};
#pragma unroll
    for (int j = 0; j < KSTEPS; ++j) {
        // A fragment (16x4 fp32): lane half selects K-pair {4j+2h, 4j+2h+1}
        v2f a; a.x = h[2 * j]; a.y = h[2 * j + 1];
        const int kk = 2 * j + half;                 // k-pair row in prepped B
        v2f b0 = bp[kk * PAD +  0 + l16];
        v2f b1 = bp[kk * PAD + 16 + l16];
        v2f b2 = bp[kk * PAD + 32 + l16];
        c0 = __builtin_amdgcn_wmma_f32_16x16x4_f32(false, a, false, b0, (short)0, c0, false, false);
        c1 = __builtin_amdgcn_wmma_f32_16x16x4_f32(false, a, false, b1, (short)0, c1, false, false);
        c2 = __builtin_amdgcn_wmma_f32_16x16x4_f32(false, a, false, b2, (short)0, c2, false, false);
    }
    // C/D layout: VGPR v holds row M = v + 8*half, col = 16*t + l16.
    // Apply bias + sin and stage to LDS in row-major [row][col].
#pragma unroll
    for (int v = 0; v < 8; ++v) {
        const int m = v + 8 * half;
        stg[m * STG_STRIDE +  0 + l16] = hw_sin(c0[v] + bias[ 0 + l16]);
        stg[m * STG_STRIDE + 16 + l16] = hw_sin(c1[v] + bias[16 + l16]);
        stg[m * STG_STRIDE + 32 + l16] = hw_sin(c2[v] + bias[32 + l16]);
    }
    // Same-wave LDS ops are in-order; wait before consuming loads.
    asm volatile("s_wait_dscnt 0x0" ::: "memory");
#pragma unroll
    for (int j = 0; j < KSTEPS; ++j) {
        // reload in A-fragment layout: row = l16, cols {4j+2h, 4j+2h+1}
        v2f t = *(const v2f*)&stg[l16 * STG_STRIDE + 4 * j + 2 * half];
        h[2 * j] = t.x; h[2 * j + 1] = t.y;
    }
    asm volatile("s_wait_dscnt 0x0" ::: "memory");
}

__global__ __launch_bounds__(256)
void siren_hyper_kernel(const float* __restrict__ x,
                        const float* __restrict__ params,
                        float* __restrict__ out, int N)
{
    __shared__ float praw[NPARAMS + 7];
    __shared__ v2f   bp2[24 * PAD];        // layer-2 B (W^T), k-pair packed, zero padded
    __shared__ v2f   bp3[24 * PAD];        // layer-3 B
    __shared__ v2f   w1p[PAD];             // (W1[c][0], W1[c][1]) padded
    __shared__ float b1p[PAD], b2p[PAD], b3p[PAD], w4p[PAD];
    __shared__ __align__(16) float stage[WAVES * 16 * STG_STRIDE];

    const int tid  = threadIdx.x;
    const int wave = tid >> 5;
    const int lane = tid & 31;
    const int half = lane >> 4;
    const int l16  = lane & 15;
    const int b    = blockIdx.y;
    const int rowbase = blockIdx.x * ROWS_PER_BLOCK + wave * 16;

    // ---- stage raw params for this batch ----
    for (int i = tid; i < NPARAMS; i += 256) praw[i] = params[b * NPARAMS + i];
    __syncthreads();

    // ---- build padded / pre-transposed operands ----
    for (int i = tid; i < PAD; i += 256) {
        v2f w1; w1.x = 0.f; w1.y = 0.f;
        float bb1 = 0.f, bb2 = 0.f, bb3 = 0.f, w4 = 0.f;
        if (i < HID) {
            w1.x = praw[OFF_W1 + i * 2 + 0];
            w1.y = praw[OFF_W1 + i * 2 + 1];
            bb1 = praw[OFF_B1 + i];
            bb2 = praw[OFF_B2 + i];
            bb3 = praw[OFF_B3 + i];
            w4  = praw[OFF_W4 + i];
        }
        w1p[i] = w1; b1p[i] = bb1; b2p[i] = bb2; b3p[i] = bb3; w4p[i] = w4;
    }
    for (int i = tid; i < 24 * PAD; i += 256) {
        const int n  = i % PAD;        // output feature (B column)
        const int k  = (i / PAD) * 2;  // input feature pair base (B row)
        v2f v; v.x = 0.f; v.y = 0.f;
        if (n < HID && k < HID) {      // HID even -> k+1 < HID too
            v.x = praw[OFF_W2 + n * HID + k + 0];
            v.y = praw[OFF_W2 + n * HID + k + 1];
        }
        bp2[i] = v;
        v.x = 0.f; v.y = 0.f;
        if (n < HID && k < HID) {
            v.x = praw[OFF_W3 + n * HID + k + 0];
            v.y = praw[OFF_W3 + n * HID + k + 1];
        }
        bp3[i] = v;
    }
    __syncthreads();

    // ---- layer 1 (K=2) straight into A-fragment layout ----
    int xrow = rowbase + l16; if (xrow >= N) xrow = N - 1;
    const v2f xm = ((const v2f*)x)[xrow];
    float h[24];
#pragma unroll
    for (int e = 0; e < 24; ++e) {
        const int c = 4 * (e >> 1) + 2 * half + (e & 1);   // zero-padded cols -> sin(0)=0
        const v2f w = w1p[c];
        h[e] = hw_sin(fmaf(xm.x, w.x, fmaf(xm.y, w.y, b1p[c])));
    }

    // ---- layers 2 and 3: fp32 WMMA ----
    hidden_layer_wmma(h, bp2, b2p, &stage[wave * 16 * STG_STRIDE], half, l16);
    hidden_layer_wmma(h, bp3, b3p, &stage[wave * 16 * STG_STRIDE], half, l16);

    // ---- layer 4: dot over lane's 24 columns, reduce across lane halves ----
    float partial = 0.f;
#pragma unroll
    for (int e = 0; e < 24; ++e) {
        const int c = 4 * (e >> 1) + 2 * half + (e & 1);
        partial = fmaf(h[e], w4p[c], partial);
    }
    partial += __shfl_xor(partial, 16, 32);

    if (half == 0) {
        const int row = rowbase + l16;
        if (row < N) {
            const float z  = partial + praw[OFF_B4];
            const float sp = fmaxf(z, 0.f) + log1pf(expf(-fabsf(z)));  // stable softplus
            out[(size_t)b * (size_t)N + row] = sp;
        }
    }
}

extern "C" void kernel_launch(void* const* d_in, const int* in_sizes, int n_in,
                              void* d_out, int out_size, void* d_ws, size_t ws_size,
                              hipStream_t stream) {
    const float* x      = (const float*)d_in[0];
    const float* params = (const float*)d_in[1];
    float* out          = (float*)d_out;
    const int N = in_sizes[0] / IN_F;        // 65536
    const int B = in_sizes[1] / NPARAMS;     // 32
    dim3 grid((N + ROWS_PER_BLOCK - 1) / ROWS_PER_BLOCK, B);
    siren_hyper_kernel<<<grid, 256, 0, stream>>>(x, params, out, N);
}